// DeformableConvBlock_43765716746649
// MI455X (gfx1250) — compile-verified
//
#include <hip/hip_runtime.h>
#include <math.h>

typedef float v2f __attribute__((ext_vector_type(2)));
typedef float v8f __attribute__((ext_vector_type(8)));

#define BB 4
#define CC 64
#define OO 64
#define HH 128
#define WW 128
#define KTAPS 9
#define KDIM (KTAPS * CC)   // 576, kdim = k*64 + c
#define HW (HH * WW)

// ---------------- Kernel 0: weight transpose  w[o][c][k] -> wT[o][k*64+c] ---
__global__ __launch_bounds__(256) void wtrans_kernel(
    const float* __restrict__ dw, float* __restrict__ wT) {
  int i = blockIdx.x * 256 + threadIdx.x;
  if (i >= OO * KDIM) return;
  int o = i / KDIM, kd = i - o * KDIM;
  int k = kd >> 6, c = kd & 63;
  wT[i] = dw[(o * CC + c) * KTAPS + k];
}

// ---------------- Kernel 1: offset conv (1 -> 18 channels, 3x3, pad 1) ------
__global__ __launch_bounds__(256) void offset_conv_kernel(
    const float* __restrict__ ele, const float* __restrict__ ow,
    const float* __restrict__ ob, float* __restrict__ off) {
  __shared__ float ws[18 * 9];
  __shared__ float bs[18];
  int tid = threadIdx.x;
  if (tid < 18 * 9) ws[tid] = ow[tid];
  if (tid < 18) bs[tid] = ob[tid];
  __syncthreads();
  int g = blockIdx.x * 256 + tid;           // b*HW + h*W + w
  int b = g / HW, hw = g - b * HW;
  int h = hw / WW, w = hw - h * WW;
  const float* eb = ele + (size_t)b * HW;
  float e[9];
#pragma unroll
  for (int dy = 0; dy < 3; ++dy)
#pragma unroll
    for (int dx = 0; dx < 3; ++dx) {
      int yy = h + dy - 1, xx = w + dx - 1;
      bool v = (yy >= 0) && (yy < HH) && (xx >= 0) && (xx < WW);
      e[dy * 3 + dx] = v ? eb[yy * WW + xx] : 0.0f;
    }
  for (int j = 0; j < 18; ++j) {
    float s = bs[j];
#pragma unroll
    for (int t = 0; t < 9; ++t) s += e[t] * ws[j * 9 + t];
    off[((size_t)b * 18 + j) * HW + hw] = s;
  }
}

// ---------------- Kernel 2: mask conv (64 -> 9 channels, 3x3) + sigmoid -----
__global__ __launch_bounds__(256) void mask_conv_kernel(
    const float* __restrict__ x, const float* __restrict__ mw,
    const float* __restrict__ mb, float* __restrict__ mask) {
  __shared__ float ws[9 * CC * 9];   // [j][c][tap] = 5184 floats (20.7 KB)
  __shared__ float bs[9];
  int tid = threadIdx.x;
  for (int i = tid; i < 9 * CC * 9; i += 256) ws[i] = mw[i];
  if (tid < 9) bs[tid] = mb[tid];
  __syncthreads();
  int g = blockIdx.x * 256 + tid;
  int b = g / HW, hw = g - b * HW;
  int h = hw / WW, w = hw - h * WW;
  float s[9];
#pragma unroll
  for (int j = 0; j < 9; ++j) s[j] = bs[j];
  for (int c = 0; c < CC; ++c) {
    const float* xc = x + ((size_t)b * CC + c) * HW;
    float e[9];
#pragma unroll
    for (int dy = 0; dy < 3; ++dy)
#pragma unroll
      for (int dx = 0; dx < 3; ++dx) {
        int yy = h + dy - 1, xx = w + dx - 1;
        bool v = (yy >= 0) && (yy < HH) && (xx >= 0) && (xx < WW);
        e[dy * 3 + dx] = v ? xc[yy * WW + xx] : 0.0f;
      }
#pragma unroll
    for (int j = 0; j < 9; ++j) {
      const float* wj = &ws[(j * CC + c) * 9];
#pragma unroll
      for (int t = 0; t < 9; ++t) s[j] += e[t] * wj[t];
    }
  }
#pragma unroll
  for (int j = 0; j < 9; ++j)
    mask[((size_t)b * 9 + j) * HW + hw] = 1.0f / (1.0f + __expf(-s[j]));
}

// --- async Global->LDS stage of one tap's weight slice (64 o x 64 c, 16 KB).
// Each of the 128 threads copies 8 x 16B with GLOBAL_LOAD_ASYNC_TO_LDS_B128
// (tracked by ASYNCcnt). LDS byte offset = low 32 bits of the generic addr.
__device__ __forceinline__ void stage_tap_async(const float* __restrict__ wT,
                                                float* dstLds, int k,
                                                int tid) {
  int o = tid >> 1, seg = tid & 1;
  const float* g = wT + (size_t)o * KDIM + k * CC + seg * 32;
  float* d = dstLds + o * CC + seg * 32;
#pragma unroll
  for (int j = 0; j < 8; ++j) {
    unsigned loff = (unsigned)(uintptr_t)(d + j * 4);
    const float* ga = g + j * 4;
    asm volatile("global_load_async_to_lds_b128 %0, %1, off"
                 :
                 : "v"(loff), "v"(ga)
                 : "memory");
  }
}

__device__ __forceinline__ void wait_async_all() {
  asm volatile("s_wait_asynccnt 0x0" ::: "memory");
}

// ---------------- Kernel 3: fused bilinear gather + WMMA GEMM + ReLU --------
// Block = one (b, h) row: 128 pixels, 4 waves x 32 pixels.
// Wave tiles: 4 M-tiles (64 out channels) x 2 N-tiles (32 pixels), K = 576 in
// chunks of 4 via v_wmma_f32_16x16x4_f32 (kdim = k*64 + c; chunk = 4 channels of one tap).
// Weights double-buffered in LDS via async Global->LDS copies, one tap ahead.
__global__ __launch_bounds__(128) void deform_wmma_kernel(
    const float* __restrict__ x, const float* __restrict__ wT,
    const float* __restrict__ offb, const float* __restrict__ maskb,
    float* __restrict__ out) {
  __shared__ __attribute__((aligned(16))) float wbuf[2][CC * CC];  // 2x16 KB
  __shared__ float4 meta[128 * KTAPS];  // {y0(int bits), x0(int bits), wy, wx}
  __shared__ float  mlds[128 * KTAPS];  // sigmoid mask per (pixel, tap)

  const int bh = blockIdx.x;
  const int b = bh / HH, h = bh - b * HH;
  const int tid = threadIdx.x;

  // Kick off async staging of tap 0 weights before metadata math.
  stage_tap_async(wT, &wbuf[0][0], 0, tid);

  // --- stage per-pixel, per-tap sampling metadata (one pixel per thread) ---
  {
    const int p = tid;
    const float* offp = offb + (size_t)b * 18 * HW + h * WW + p;
    const float* mp   = maskb + (size_t)b * KTAPS * HW + h * WW + p;
#pragma unroll
    for (int k = 0; k < KTAPS; ++k) {
      float oy = offp[(2 * k) * HW];
      float ox = offp[(2 * k + 1) * HW];
      float py = oy + (float)h + (float)(k / 3) - 1.0f;
      float px = ox + (float)p + (float)(k % 3) - 1.0f;
      float y0f = floorf(py), x0f = floorf(px);
      float4 m;
      m.x = __int_as_float((int)y0f);
      m.y = __int_as_float((int)x0f);
      m.z = py - y0f;
      m.w = px - x0f;
      meta[p * KTAPS + k] = m;
      mlds[p * KTAPS + k] = mp[k * HW];
    }
  }
  wait_async_all();
  __syncthreads();

  const int wave = tid >> 5, lane = tid & 31;
  const int n = lane & 15, half = lane >> 4;
  const int w0 = wave * 32;

  v8f acc[4][2];
#pragma unroll
  for (int m = 0; m < 4; ++m)
#pragma unroll
    for (int t = 0; t < 2; ++t) acc[m][t] = (v8f)0.0f;

  const float* xb = x + (size_t)b * CC * HW;

  for (int k = 0; k < KTAPS; ++k) {
    const float* wl = &wbuf[k & 1][0];
    // Prefetch next tap's weight slice into the other LDS buffer.
    if (k + 1 < KTAPS) stage_tap_async(wT, &wbuf[(k + 1) & 1][0], k + 1, tid);

    // Hoist bilinear corner indices & combined weights once per tap.
    int   idx[2][4];
    float cw[2][4];
#pragma unroll
    for (int t = 0; t < 2; ++t) {
      int p = w0 + t * 16 + n;
      float4 m = meta[p * KTAPS + k];
      int y0 = __float_as_int(m.x), x0 = __float_as_int(m.y);
      float wy = m.z, wx = m.w;
      float mv = mlds[p * KTAPS + k];
      int y1 = y0 + 1, x1 = x0 + 1;
      bool vy0 = (y0 >= 0) && (y0 < HH), vy1 = (y1 >= 0) && (y1 < HH);
      bool vx0 = (x0 >= 0) && (x0 < WW), vx1 = (x1 >= 0) && (x1 < WW);
      int yc0 = min(max(y0, 0), HH - 1), yc1 = min(max(y1, 0), HH - 1);
      int xc0 = min(max(x0, 0), WW - 1), xc1 = min(max(x1, 0), WW - 1);
      idx[t][0] = yc0 * WW + xc0;
      idx[t][1] = yc0 * WW + xc1;
      idx[t][2] = yc1 * WW + xc0;
      idx[t][3] = yc1 * WW + xc1;
      float wy0 = 1.0f - wy, wx0 = 1.0f - wx;
      cw[t][0] = (vy0 && vx0) ? wy0 * wx0 * mv : 0.0f;
      cw[t][1] = (vy0 && vx1) ? wy0 * wx  * mv : 0.0f;
      cw[t][2] = (vy1 && vx0) ? wy  * wx0 * mv : 0.0f;
      cw[t][3] = (vy1 && vx1) ? wy  * wx  * mv : 0.0f;
    }

    for (int c0 = 0; c0 < CC; c0 += 4) {
      // A fragments from LDS: 16x4 f32 layout; lane n -> row o = m*16+n,
      // half-wave selects K values {0,1} vs {2,3}.
      v2f afrag[4];
#pragma unroll
      for (int m = 0; m < 4; ++m) {
        int o = m * 16 + n;
        afrag[m] = *(const v2f*)&wl[o * CC + c0 + half * 2];
      }

      // B fragments: 4x16 f32; lane col = pixel n, half selects K rows.
      v2f bfrag[2];
#pragma unroll
      for (int t = 0; t < 2; ++t) {
        int c = c0 + half * 2;
        const float* xc0p = xb + (size_t)c * HW;
        const float* xc1p = xc0p + HW;
        float s0 = cw[t][0] * xc0p[idx[t][0]] + cw[t][1] * xc0p[idx[t][1]] +
                   cw[t][2] * xc0p[idx[t][2]] + cw[t][3] * xc0p[idx[t][3]];
        float s1 = cw[t][0] * xc1p[idx[t][0]] + cw[t][1] * xc1p[idx[t][1]] +
                   cw[t][2] * xc1p[idx[t][2]] + cw[t][3] * xc1p[idx[t][3]];
        bfrag[t].x = s0;
        bfrag[t].y = s1;
      }

#pragma unroll
      for (int m = 0; m < 4; ++m)
#pragma unroll
        for (int t = 0; t < 2; ++t)
          acc[m][t] = __builtin_amdgcn_wmma_f32_16x16x4_f32(
              false, afrag[m], false, bfrag[t], (short)0, acc[m][t], false,
              false);
    }

    // Publish next tap's weights: this wave's async copies done + barrier.
    wait_async_all();
    __syncthreads();
  }

  // --- epilogue: ReLU + store; C/D layout: VGPR r holds rows r / r+8 --------
#pragma unroll
  for (int m = 0; m < 4; ++m)
#pragma unroll
    for (int t = 0; t < 2; ++t) {
      int wcol = w0 + t * 16 + n;
#pragma unroll
      for (int r = 0; r < 8; ++r) {
        int o = m * 16 + half * 8 + r;
        out[((size_t)(b * OO + o) * HH + h) * WW + wcol] =
            fmaxf(acc[m][t][r], 0.0f);
      }
    }
}

extern "C" void kernel_launch(void* const* d_in, const int* in_sizes, int n_in,
                              void* d_out, int out_size, void* d_ws,
                              size_t ws_size, hipStream_t stream) {
  (void)in_sizes; (void)n_in; (void)out_size; (void)ws_size;
  const float* x   = (const float*)d_in[0];
  const float* ele = (const float*)d_in[1];
  const float* ow  = (const float*)d_in[2];
  const float* ob  = (const float*)d_in[3];
  const float* mw  = (const float*)d_in[4];
  const float* mb  = (const float*)d_in[5];
  const float* dw  = (const float*)d_in[6];
  float* out = (float*)d_out;

  // workspace layout: offsets | mask | transposed weights
  float* offb  = (float*)d_ws;                       // B*18*HW
  float* maskb = offb + (size_t)BB * 18 * HW;        // B*9*HW
  float* wT    = maskb + (size_t)BB * KTAPS * HW;    // OO*KDIM

  const int npix = BB * HW;  // 65536
  wtrans_kernel<<<(OO * KDIM + 255) / 256, 256, 0, stream>>>(dw, wT);
  offset_conv_kernel<<<npix / 256, 256, 0, stream>>>(ele, ow, ob, offb);
  mask_conv_kernel<<<npix / 256, 256, 0, stream>>>(x, mw, mb, maskb);
  deform_wmma_kernel<<<BB * HH, 128, 0, stream>>>(x, wT, offb, maskb, out);
}